// RotaryPositionalEmbedding_18597208391792
// MI455X (gfx1250) — compile-verified
//
#include <hip/hip_runtime.h>
#include <math.h>

// RoPE forward, fp32, D_K = 128, consecutive-pair rotation.
// Memory-bound streaming kernel: ~134 MB HBM traffic, floor ~5.8 us @ 23.3 TB/s.
//
// Strategy: angles depend only on (position, pair) with position < 8192 and
// 64 pairs -> precompute a 4 MiB (cos,sin) table into d_ws (L2-resident on
// 192 MB L2), then the hot kernel is pure data movement:
//   per lane: NT b128 load of x, b128 table load (L2 hit), 8 FMAs, NT b128 store.

namespace {
constexpr int   kDk            = 128;
constexpr int   kPairs         = kDk / 2;      // 64
constexpr int   kMaxSeqLen     = 8192;         // reference MAX_SEQ_LEN
constexpr int   kWavesPerBlock = 8;            // 256 threads = 8 wave32
constexpr int   kBlockThreads  = kWavesPerBlock * 32;
// -log2(10000) / 64
constexpr float kNegLog2ThetaOver64 = -0.207620505930460f;
}

typedef __attribute__((ext_vector_type(4))) float v4f;
typedef __attribute__((ext_vector_type(2))) float v2f;

// ---------------------------------------------------------------------------
// Setup: table[p][j] = (cos(p*inv_freq_j), sin(p*inv_freq_j)), p<8192, j<64.
// 512K lanes, heavy sincos lives ONLY here (runs once per call, ~0.1% of time).
// ---------------------------------------------------------------------------
__global__ __launch_bounds__(256)
void rope_table_kernel(v2f* __restrict__ table)
{
    const int idx = blockIdx.x * 256 + threadIdx.x;     // [0, 8192*64)
    if (idx >= kMaxSeqLen * kPairs) return;
    const int p = idx >> 6;         // position
    const int j = idx & 63;         // pair index
    const float inv = exp2f((float)j * kNegLog2ThetaOver64);
    float s, c;
    sincosf((float)p * inv, &s, &c);    // full range reduction, fp32-accurate
    v2f cs; cs.x = c; cs.y = s;
    table[idx] = cs;
}

// ---------------------------------------------------------------------------
// Hot kernel: one wave32 per row of 128 floats. Lane L owns dims [4L, 4L+3],
// i.e. rotation pairs j0 = 2L, j1 = 2L+1. Table slice per wave = 512 B,
// coalesced, L2-resident.
// ---------------------------------------------------------------------------
__global__ __launch_bounds__(kBlockThreads)
void rope_fwd_table_kernel(const float* __restrict__ x,
                           const int*   __restrict__ pos,
                           const float* __restrict__ table,   // viewed as v4f
                           float*       __restrict__ out,
                           int n_rows)
{
    const int lane = threadIdx.x & 31;
    const int wave = threadIdx.x >> 5;
    const int row  = blockIdx.x * kWavesPerBlock + wave;
    if (row >= n_rows) return;

    const size_t base = (size_t)row * kDk;

    // gfx1250 prefetch: pull the read stream 64 rows (32 KB) ahead into L2.
    {
        const int pf_row = row + 8 * kWavesPerBlock;
        if (pf_row < n_rows) {
            __builtin_prefetch(x + (size_t)pf_row * kDk + lane * 4, /*rw=*/0, /*loc=*/1);
        }
    }

    // Single-use stream -> non-temporal load (don't evict the table from L2).
    const v4f v = __builtin_nontemporal_load(
        reinterpret_cast<const v4f*>(x + base) + lane);

    // Wave-uniform position -> broadcast load. Guaranteed < kMaxSeqLen.
    const int p = pos[row];

    // (c0, s0, c1, s1) for pairs 2*lane, 2*lane+1: contiguous 512 B per wave.
    const v4f t = reinterpret_cast<const v4f*>(table)[(size_t)p * (kPairs / 2) + lane];

    v4f r;
    r.x = t.x * v.x - t.y * v.y;    // pair j0: cos*x0 - sin*x1
    r.y = t.y * v.x + t.x * v.y;    //          sin*x0 + cos*x1
    r.z = t.z * v.z - t.w * v.w;    // pair j1
    r.w = t.w * v.z + t.z * v.w;

    __builtin_nontemporal_store(r, reinterpret_cast<v4f*>(out + base) + lane);
}

// ---------------------------------------------------------------------------
// Fallback (only if ws_size < 4 MiB): compute sincos in the hot loop.
// ---------------------------------------------------------------------------
__global__ __launch_bounds__(kBlockThreads)
void rope_fwd_direct_kernel(const float* __restrict__ x,
                            const int*   __restrict__ pos,
                            float*       __restrict__ out,
                            int n_rows)
{
    const int lane = threadIdx.x & 31;
    const int wave = threadIdx.x >> 5;
    const int row  = blockIdx.x * kWavesPerBlock + wave;
    if (row >= n_rows) return;

    const size_t base = (size_t)row * kDk;
    const v4f v = __builtin_nontemporal_load(
        reinterpret_cast<const v4f*>(x + base) + lane);
    const float p = (float)pos[row];

    const float j0   = (float)(2 * lane);
    const float inv0 = exp2f(j0 * kNegLog2ThetaOver64);
    const float inv1 = exp2f((j0 + 1.0f) * kNegLog2ThetaOver64);

    float s0, c0, s1, c1;
    sincosf(p * inv0, &s0, &c0);
    sincosf(p * inv1, &s1, &c1);

    v4f r;
    r.x = c0 * v.x - s0 * v.y;
    r.y = s0 * v.x + c0 * v.y;
    r.z = c1 * v.z - s1 * v.w;
    r.w = s1 * v.z + c1 * v.w;
    __builtin_nontemporal_store(r, reinterpret_cast<v4f*>(out + base) + lane);
}

extern "C" void kernel_launch(void* const* d_in, const int* in_sizes, int n_in,
                              void* d_out, int out_size, void* d_ws, size_t ws_size,
                              hipStream_t stream)
{
    (void)n_in; (void)out_size;

    const float* x   = (const float*)d_in[0];   // [B, S, 128] fp32
    const int*   pos = (const int*)d_in[1];     // [B, S] int32
    float*       out = (float*)d_out;           // [B, S, 128] fp32

    const int n_rows = in_sizes[1];             // B*S rows of 128 elements
    const int blocks = (n_rows + kWavesPerBlock - 1) / kWavesPerBlock;

    const size_t table_bytes = (size_t)kMaxSeqLen * kPairs * sizeof(v2f);  // 4 MiB

    if (ws_size >= table_bytes) {
        v2f* table = (v2f*)d_ws;
        const int n_entries     = kMaxSeqLen * kPairs;          // 524288
        const int table_blocks  = (n_entries + 255) / 256;
        // Same stream -> table kernel completes before the hot kernel reads it
        // (holds under graph capture as sequential nodes too).
        rope_table_kernel<<<table_blocks, 256, 0, stream>>>(table);
        rope_fwd_table_kernel<<<blocks, kBlockThreads, 0, stream>>>(
            x, pos, (const float*)table, out, n_rows);
    } else {
        rope_fwd_direct_kernel<<<blocks, kBlockThreads, 0, stream>>>(
            x, pos, out, n_rows);
    }
}